// InteractiveGallingModelV2_65111704207991
// MI455X (gfx1250) — compile-verified
//
#include <hip/hip_runtime.h>

// ---------------------------------------------------------------------------
// InteractiveGallingModelV2: B=131072 independent trajectories, N=150 serial
// steps. Memory-bound streaming scan (~630MB @ 23.3TB/s ~= 27us floor).
// Strategy: 1 thread / trajectory, double-buffered async global->LDS staging
// of the 3 noise streams (gfx1250 GLOBAL_LOAD_ASYNC_TO_LDS + s_wait_asynccnt,
// NT temporal hint), raw base-2 hardware transcendentals + native v_tanh_f32,
// non-temporal stores for the 393MB output (bigger than the 192MB L2).
// ---------------------------------------------------------------------------

#define BLK 256   // 8 wave32 per block
#define CH  6     // time-steps staged per chunk (divides N=150 -> 25 chunks)

#define AS1 __attribute__((address_space(1)))
#define AS3 __attribute__((address_space(3)))

#define LOG2E 1.44269504088896340736f

__device__ __forceinline__ void async_copy_b32(const float* g, float* l) {
#if __has_builtin(__builtin_amdgcn_global_load_async_to_lds_b32)
  // per-lane async DMA: copies 4 bytes from the global address in g to the LDS address in l; ASYNCcnt-tracked; cpol=1 -> TH_NT
  // (noise streams are read exactly once; don't pollute L2)
  __builtin_amdgcn_global_load_async_to_lds_b32(
      (AS1 int*)(unsigned long long)g,
      (AS3 int*)(unsigned)(unsigned long long)l,
      /*imm offset*/0, /*cpol TH_NT*/1);
#else
  *l = __builtin_nontemporal_load(g);   // fallback keeps compile green
#endif
}

__device__ __forceinline__ void wait_async_zero() {
#if __has_builtin(__builtin_amdgcn_s_wait_asynccnt)
  __builtin_amdgcn_s_wait_asynccnt(0);
#else
  asm volatile("s_wait_asynccnt 0" ::: "memory");
#endif
  asm volatile("" ::: "memory");  // keep LDS consumption below the wait
}

// sigmoid(x) with pre-scaled arg: expects xl2 = -x*log2(e).
// 1/(1 + 2^xl2): v_exp_f32 + v_add + v_rcp_f32
__device__ __forceinline__ float sigmoid_l2(float xl2) {
  return __builtin_amdgcn_rcpf(1.0f + __builtin_amdgcn_exp2f(xl2));
}

// x^a for x > 0: v_log_f32 + v_mul + v_exp_f32 (no powf fixup code)
__device__ __forceinline__ float pow_pos(float x, float a) {
  return __builtin_amdgcn_exp2f(a * __builtin_amdgcn_logf(x));
}

__device__ __forceinline__ float tanh_pos(float x) {
#if __has_builtin(__builtin_amdgcn_tanhf)
  return __builtin_amdgcn_tanhf(x);          // native v_tanh_f32 (gfx1250)
#else
  // x >= 0: tanh = (1-e^-2x)/(1+e^-2x), base-2 form
  float e2 = __builtin_amdgcn_exp2f(-2.0f * LOG2E * x);
  return (1.0f - e2) * __builtin_amdgcn_rcpf(1.0f + e2);
#endif
}

__global__ __launch_bounds__(BLK) void galling_scan_kernel(
    const float* __restrict__ M_init, const float* __restrict__ rho_init,
    const float* __restrict__ noise_M, const float* __restrict__ noise_mu,
    const float* __restrict__ unif_detach,
    const float* __restrict__ pT,
    const float* __restrict__ p_k_adh, const float* __restrict__ p_alpha_tau,
    const float* __restrict__ p_k_spall, const float* __restrict__ p_E_spall,
    const float* __restrict__ p_heal, const float* __restrict__ p_kappa,
    const float* __restrict__ p_M_sat, const float* __restrict__ p_rho_crit,
    const float* __restrict__ p_beta_sharp, const float* __restrict__ p_noise_lvl,
    const float* __restrict__ p_prob_detach, const float* __restrict__ p_out_noise,
    float* __restrict__ out, int B, int N)
{
  // double-buffered staging: [buf][stream][step][lane] = 2*3*CH*BLK floats = 36KB
  __shared__ float sbuf[2 * 3 * CH * BLK];

  const int tid = (int)threadIdx.x;
  const int gid = (int)blockIdx.x * BLK + tid;
  if (gid >= B) return;

  // ---- uniform scalar params (loaded once, folded into derived constants) ----
  const float T           = pT[0];
  const float k_adh       = p_k_adh[0];
  const float alpha_tau   = p_alpha_tau[0];
  const float k_spall     = p_k_spall[0];
  const float E_spall     = p_E_spall[0];
  const float heal_exp    = p_heal[0];
  const float kappa       = p_kappa[0];
  const float M_sat       = p_M_sat[0];
  const float rho_crit    = p_rho_crit[0];
  const float bs          = p_beta_sharp[0];
  const float noise_lvl   = p_noise_lvl[0];
  const float prob_detach = p_prob_detach[0];
  const float out_noise   = p_out_noise[0];

  const float dT        = T - 165.0f;                               // T_REF
  const float kg_coef   = k_adh * __builtin_amdgcn_exp2f(LOG2E * 0.2f * dT);
  const float k_removal = k_spall * __builtin_amdgcn_exp2f(-LOG2E * E_spall * dT);
  const float p_det     = prob_detach * __builtin_amdgcn_exp2f(-LOG2E * 0.5f * dT);
  const float inv_Msat  = 1.0f / M_sat;
  const float bsl2      = bs * LOG2E;    // sigmoid(bs*(x-rc)) = sigmoid_l2(bsl2*(rc-x))
  const float MU_LOW_C  = 0.15f;
  const float MU_SPAN   = 1.0f - 0.15f;                             // MU_HIGH - MU_LOW

  // ---- carried state. beta at step start == beta2 of previous step,
  //      so carry beta instead of recomputing sigmoid(rho) each step. ----
  float M    = M_init[gid];
  float beta = sigmoid_l2(bsl2 * (rho_crit - rho_init[gid]));
  float mu   = MU_LOW_C;

  const size_t NB  = (size_t)N * (size_t)B;
  const int    nch = (N + CH - 1) / CH;

  auto issue_chunk = [&](int c) {
    const int t0 = c * CH;
    float* base = &sbuf[(c & 1) * (3 * CH * BLK)];
#pragma unroll
    for (int s = 0; s < CH; ++s) {
      if (t0 + s >= N) break;                             // wave-uniform
      const size_t off = (size_t)(t0 + s) * (size_t)B + (size_t)gid;
      async_copy_b32(noise_M     + off, base + (0 * CH + s) * BLK + tid);
      async_copy_b32(noise_mu    + off, base + (1 * CH + s) * BLK + tid);
      async_copy_b32(unif_detach + off, base + (2 * CH + s) * BLK + tid);
    }
  };

  issue_chunk(0);   // prime the pipeline

  for (int c = 0; c < nch; ++c) {
    wait_async_zero();                                    // chunk c resident
    if (c + 1 < nch) issue_chunk(c + 1);                  // overlap next DMA

    const float* rb = &sbuf[(c & 1) * (3 * CH * BLK)];
#pragma unroll
    for (int s = 0; s < CH; ++s) {
      const int t = c * CH + s;
      if (t >= N) break;

      const float nM  = rb[(0 * CH + s) * BLK + tid];
      const float nmu = rb[(1 * CH + s) * BLK + tid];
      const float u   = rb[(2 * CH + s) * BLK + tid];

      const float k_growth = kg_coef * pow_pos(mu, alpha_tau);
      const float comp     = 1.0f + kappa * beta * (1.0f - beta);
      const float heal     = pow_pos(M + 1e-6f, heal_exp);
      const float dM       = comp * (k_growth - k_removal * heal) + nM * noise_lvl;

      float Mn = fmaxf(M + dM, 0.0f);
      Mn *= (u < p_det) ? 0.5f : 1.0f;                    // stochastic detach

      const float rhon  = tanh_pos(Mn * inv_Msat);
      const float b2    = sigmoid_l2(bsl2 * (rho_crit - rhon));
      float       mun   = fmaf(b2, MU_SPAN, MU_LOW_C) + nmu * out_noise;
      mun               = fminf(fmaxf(mun, 0.1f), 1.3f);
      const float comp2 = 1.0f + kappa * b2 * (1.0f - b2);

      // out[5][N][B]: 0=mu, 1=rho, 2=M, 3=beta2, 4=comp2  (streamed -> NT)
      const size_t o = (size_t)t * (size_t)B + (size_t)gid;
      __builtin_nontemporal_store(mun,   out + 0 * NB + o);
      __builtin_nontemporal_store(rhon,  out + 1 * NB + o);
      __builtin_nontemporal_store(Mn,    out + 2 * NB + o);
      __builtin_nontemporal_store(b2,    out + 3 * NB + o);
      __builtin_nontemporal_store(comp2, out + 4 * NB + o);

      M = Mn; mu = mun; beta = b2;                        // carry
    }
  }
}

extern "C" void kernel_launch(void* const* d_in, const int* in_sizes, int n_in,
                              void* d_out, int out_size, void* d_ws, size_t ws_size,
                              hipStream_t stream) {
  (void)n_in; (void)out_size; (void)d_ws; (void)ws_size;

  const float* M_init      = (const float*)d_in[0];
  const float* rho_init    = (const float*)d_in[1];
  const float* noise_M     = (const float*)d_in[2];
  const float* noise_mu    = (const float*)d_in[3];
  const float* unif_detach = (const float*)d_in[4];
  const float* T           = (const float*)d_in[5];
  const float* k_adh       = (const float*)d_in[6];
  const float* alpha_tau   = (const float*)d_in[7];
  const float* k_spall     = (const float*)d_in[8];
  const float* E_spall     = (const float*)d_in[9];
  const float* heal        = (const float*)d_in[10];
  const float* kappa       = (const float*)d_in[11];
  const float* M_sat       = (const float*)d_in[12];
  const float* rho_crit    = (const float*)d_in[13];
  const float* beta_sharp  = (const float*)d_in[14];
  const float* noise_lvl   = (const float*)d_in[15];
  const float* prob_detach = (const float*)d_in[16];
  const float* out_noise   = (const float*)d_in[17];

  const int B = in_sizes[0];
  const int N = in_sizes[2] / B;   // noise_M is [N, B]

  dim3 grid((B + BLK - 1) / BLK), block(BLK);
  galling_scan_kernel<<<grid, block, 0, stream>>>(
      M_init, rho_init, noise_M, noise_mu, unif_detach,
      T, k_adh, alpha_tau, k_spall, E_spall, heal, kappa,
      M_sat, rho_crit, beta_sharp, noise_lvl, prob_detach, out_noise,
      (float*)d_out, B, N);
}